// ConeNet_82867099009867
// MI455X (gfx1250) — compile-verified
//
#include <hip/hip_runtime.h>
#include <hip/hip_bf16.h>
#include <math.h>

typedef __attribute__((ext_vector_type(2))) float v2f;
typedef __attribute__((ext_vector_type(8))) float v8f;

#define NPTS 524288
#define KCONES 32
#define DIM 512
#define VOXN 32768
#define ENC_BLOCKS 256
#define ENC_ROWS (VOXN / ENC_BLOCKS)   // 128

// ---------------------------------------------------------------------------
// Stage 1: encoder GEMV partials. Block b handles rows [b*128, b*128+128) of
// W_enc for all 512 columns. partial[b*512 + j] = sum_i v[i]*W[i][j].
// Deterministic (no atomics); partials live in d_out's SDF region, which is
// fully overwritten by the SDF kernel afterwards.
// ---------------------------------------------------------------------------
__global__ __launch_bounds__(256) void enc_partial(const float* __restrict__ vox,
                                                   const float* __restrict__ W,
                                                   float* __restrict__ partial) {
    __shared__ float sv[ENC_ROWS];
    const int b = blockIdx.x;
    const int t = threadIdx.x;
    const int row0 = b * ENC_ROWS;
    for (int i = t; i < ENC_ROWS; i += 256) sv[i] = vox[row0 + i];
    __syncthreads();
    float a0 = 0.f, a1 = 0.f;
    const float* Wp = W + (size_t)row0 * DIM;
    #pragma unroll 4
    for (int i = 0; i < ENC_ROWS; ++i) {
        const float v = sv[i];
        a0 = fmaf(v, Wp[(size_t)i * DIM + t], a0);
        a1 = fmaf(v, Wp[(size_t)i * DIM + t + 256], a1);
    }
    partial[(size_t)b * DIM + t]       = a0;
    partial[(size_t)b * DIM + t + 256] = a1;
}

// ---------------------------------------------------------------------------
// Stage 2: reduce partials, add bias, ReLU -> feat[512]
// ---------------------------------------------------------------------------
__global__ __launch_bounds__(256) void enc_reduce(const float* __restrict__ partial,
                                                  const float* __restrict__ b_enc,
                                                  float* __restrict__ feat) {
    const int j = blockIdx.x * 256 + threadIdx.x;
    float s = b_enc[j];
    for (int b = 0; b < ENC_BLOCKS; ++b) s += partial[(size_t)b * DIM + j];
    feat[j] = fmaxf(s, 0.f);
}

// ---------------------------------------------------------------------------
// Stage 3: dec = relu(feat @ W_dec + b_dec), 512x512 GEMV, one block.
// ---------------------------------------------------------------------------
__global__ __launch_bounds__(512) void dec_kernel(const float* __restrict__ feat,
                                                  const float* __restrict__ W_dec,
                                                  const float* __restrict__ b_dec,
                                                  float* __restrict__ dec) {
    __shared__ float sf[DIM];
    const int t = threadIdx.x;
    sf[t] = feat[t];
    __syncthreads();
    float s = b_dec[t];
    #pragma unroll 4
    for (int i = 0; i < DIM; ++i) s = fmaf(sf[i], W_dec[(size_t)i * DIM + t], s);
    dec[t] = fmaxf(s, 0.f);
}

// ---------------------------------------------------------------------------
// Stage 4: cone_params = sigmoid(dec @ W_map + b_map)*mult + add, plus per-cone
// precompute for the SDF WMMA kernel. cone[] layout (stride 12 floats):
//  [0..2]=axis, [3]=-c.a, [4..6]=-2c, [7]=|c|^2, [8]=r, [9]=h, [10]=1/(h+eps)
// ---------------------------------------------------------------------------
__global__ __launch_bounds__(256) void map_kernel(const float* __restrict__ dec,
                                                  const float* __restrict__ W_map,
                                                  const float* __restrict__ b_map,
                                                  float* __restrict__ cp_out,
                                                  float* __restrict__ cone) {
    __shared__ float sd[DIM];
    __shared__ float scp[KCONES * 8];
    const int t = threadIdx.x;
    sd[t]       = dec[t];
    sd[t + 256] = dec[t + 256];
    __syncthreads();
    float s = b_map[t];
    #pragma unroll 4
    for (int i = 0; i < DIM; ++i) s = fmaf(sd[i], W_map[(size_t)i * (KCONES * 8) + t], s);
    const float mult[8] = {1.f, 1.f, 1.f, 0.01f, 0.03f, 1.7f, 1.7f, 1.7f};
    const float addr[8] = {-0.8f, -0.8f, -0.8f, 0.05f, 0.1f, -1.f, -1.f, -1.f};
    const float sig = 1.0f / (1.0f + __expf(-s));
    const int c = t & 7;
    const float p = sig * mult[c] + addr[c];
    cp_out[t] = p;          // cone_params part of the output tuple
    scp[t] = p;
    __syncthreads();
    if (t < KCONES) {
        const float* pp = scp + t * 8;
        const float cx = pp[0], cy = pp[1], cz = pp[2];
        const float r = pp[3], h = pp[4];
        const float ox = pp[5], oy = pp[6], oz = pp[7];
        const float n = sqrtf(ox * ox + oy * oy + oz * oz) + 1e-8f;
        const float ax = ox / n, ay = oy / n, az = oz / n;
        float* o = cone + t * 12;
        o[0] = ax; o[1] = ay; o[2] = az;
        o[3] = -(cx * ax + cy * ay + cz * az);
        o[4] = -2.f * cx; o[5] = -2.f * cy; o[6] = -2.f * cz;
        o[7] = cx * cx + cy * cy + cz * cz;
        o[8] = r; o[9] = h; o[10] = 1.f / (h + 1e-8f); o[11] = 0.f;
    }
}

// ---------------------------------------------------------------------------
// Stage 5: SDF via V_WMMA_F32_16X16X4_F32.
//   proj  = WMMA( A=(qx,qy,qz,1),      B=(ax,ay,az,-c.a) )
//   dist2 = WMMA( A=(qx,qy,qz,|q|^2),  B=(-2cx,-2cy,-2cz,1) ) + |c|^2
//   perp  = sqrt(max(dist2 - proj^2, 0))
//   sdf   = max(perp - r*(1 - proj/(h+eps)), |proj| - h)
// Each wave: 16 points x 32 cones (2 column tiles), 4 WMMAs.
// ---------------------------------------------------------------------------
__global__ __launch_bounds__(256) void sdf_kernel(const float* __restrict__ q,
                                                  const float* __restrict__ cone,
                                                  float* __restrict__ out) {
    const int lane = threadIdx.x & 31;
    const int wave = blockIdx.x * (256 >> 5) + (threadIdx.x >> 5);
    const int base = wave * 16;                 // first query point of tile
    const int m    = lane & 15;
    const bool hi  = lane >= 16;

    // Each lane loads its point (lanes l and l+16 share point m).
    const float qx = q[(size_t)(base + m) * 3 + 0];
    const float qy = q[(size_t)(base + m) * 3 + 1];
    const float qz = q[(size_t)(base + m) * 3 + 2];
    const float qq = qx * qx + qy * qy + qz * qz;

    // A-matrix layout: lanes 0-15 hold (K=0,K=1), lanes 16-31 hold (K=2,K=3).
    v2f A_proj, A_dist;
    A_proj.x = hi ? qz : qx;
    A_proj.y = hi ? 1.0f : qy;
    A_dist.x = hi ? qz : qx;
    A_dist.y = hi ? qq : qy;

    #pragma unroll
    for (int tile = 0; tile < 2; ++tile) {
        const float* cp = cone + (size_t)(tile * 16 + m) * 12;
        v2f Bp, Bd;
        Bp.x = hi ? cp[2] : cp[0];   // az : ax
        Bp.y = hi ? cp[3] : cp[1];   // -c.a : ay
        Bd.x = hi ? cp[6] : cp[4];   // -2cz : -2cx
        Bd.y = hi ? 1.0f : cp[5];    //  1   : -2cy
        const float cc = cp[7], r = cp[8], h = cp[9], ih = cp[10];

        v8f zero = {};
        v8f proj = __builtin_amdgcn_wmma_f32_16x16x4_f32(
            false, A_proj, false, Bp, (short)0, zero, false, false);
        v8f d2 = __builtin_amdgcn_wmma_f32_16x16x4_f32(
            false, A_dist, false, Bd, (short)0, zero, false, false);

        // Epilogue: element (M = rI + 8*hi, N = m) of this 16x16 tile.
        float* orow = out + (size_t)(base + (hi ? 8 : 0)) * KCONES + tile * 16 + m;
        #pragma unroll
        for (int rI = 0; rI < 8; ++rI) {
            const float pj    = proj[rI];
            const float dist2 = d2[rI] + cc;
            const float perp  = sqrtf(fmaxf(dist2 - pj * pj, 0.0f));
            const float slant = perp - r * (1.0f - pj * ih);
            const float cap   = fabsf(pj) - h;
            orow[(size_t)rI * KCONES] = fmaxf(slant, cap);
        }
    }
}

// ---------------------------------------------------------------------------
extern "C" void kernel_launch(void* const* d_in, const int* in_sizes, int n_in,
                              void* d_out, int out_size, void* d_ws, size_t ws_size,
                              hipStream_t stream) {
    const float* voxel  = (const float*)d_in[0];
    const float* qpts   = (const float*)d_in[1];
    // d_in[2] = initial_centers: unused by the reference
    const float* W_enc  = (const float*)d_in[3];
    const float* b_enc  = (const float*)d_in[4];
    const float* W_dec  = (const float*)d_in[5];
    const float* b_dec  = (const float*)d_in[6];
    const float* W_map  = (const float*)d_in[7];
    const float* b_map  = (const float*)d_in[8];

    float* out  = (float*)d_out;
    float* ws   = (float*)d_ws;
    float* feat = ws;                 // 512 floats
    float* dec  = ws + 512;           // 512 floats
    float* cone = ws + 1024;          // 384 floats (32 cones x 12)

    // Encoder partials live in d_out's SDF region (ENC_BLOCKS*512 = 131072
    // floats << N*K); fully overwritten by sdf_kernel afterwards.
    float* partial = out;

    enc_partial<<<ENC_BLOCKS, 256, 0, stream>>>(voxel, W_enc, partial);
    enc_reduce<<<2, 256, 0, stream>>>(partial, b_enc, feat);
    dec_kernel<<<1, 512, 0, stream>>>(feat, W_dec, b_dec, dec);
    map_kernel<<<1, 256, 0, stream>>>(dec, W_map, b_map,
                                      out + (size_t)NPTS * KCONES, cone);
    // N/16 = 32768 waves, 8 waves per 256-thread block -> 4096 blocks
    sdf_kernel<<<NPTS / (16 * 8), 256, 0, stream>>>(qpts, cone, out);
}